// EPGSimulation_63694365000169
// MI455X (gfx1250) — compile-verified
//
#include <hip/hip_runtime.h>
#include <hip/hip_bf16.h>

#define N_PULSES 64
#define BATCH    16384
#define N_STATES 21
#define TPB      32                      // single wave32 per workgroup -> barriers are S_NOP
#define EPT      (N_STATES * 5)          // 105 floats per thread per pulse
#define CHUNK    (TPB * EPT)             // 3360 floats per block per pulse (13440 B, contiguous in out)
#define CHUNK4   (CHUNK / 4)             // 840 x 16B vectors

#if defined(__has_builtin)
#  if __has_builtin(__builtin_amdgcn_global_store_async_from_lds_b128)
#    define EPG_ASYNC 1
#  endif
#endif

// pointer types matching the builtin's parameters: pointer to 16B int vector,
// in global (AS1) / LDS (AS3) address spaces
typedef int epg_v4i __attribute__((vector_size(16)));
typedef __attribute__((address_space(1))) epg_v4i* epg_gptr;
typedef __attribute__((address_space(3))) epg_v4i* epg_lptr;

__device__ __forceinline__ void epg_wait_dscnt0() {
    asm volatile("s_wait_dscnt 0" ::: "memory");
}
#if defined(EPG_ASYNC)
__device__ __forceinline__ void epg_wait_async_le27() {
    // allow previous pulse's 27 outstanding async stores; pulse p-2's buffer is free
    asm volatile("s_wait_asynccnt 27" ::: "memory");
}
#endif

__global__ __launch_bounds__(TPB) void epg_sim_kernel(
    const float* __restrict__ flip, const float* __restrict__ phases,
    const float* __restrict__ T1v, const float* __restrict__ T2v,
    const float* __restrict__ B0v, const float* __restrict__ B1v,
    const int* __restrict__ TRp, float* __restrict__ out)
{
    __shared__ __align__(16) float lds[2][CHUNK];   // double-buffered staging (26,880 B)

    const int t = threadIdx.x;
    const int b = blockIdx.x * TPB + t;

    const float TR = (float)(*TRp);
    const float T1 = T1v[b], T2 = T2v[b], B0 = B0v[b], B1 = B1v[b];

    const float E1     = __expf(-TR / T1);
    const float E2     = __expf(-TR / T2);
    const float regrow = 1.0f - E1;
    const float phi    = 2.0f * 3.14159265358979323846f * B0 * TR * 0.001f;
    const float b0r    = __cosf(phi);
    const float b0i    = __sinf(phi);

    // state: registers (fully unrolled constant indexing -> SROA to VGPRs)
    float FpR[N_STATES], FpI[N_STATES], FmR[N_STATES], FmI[N_STATES], Zs[N_STATES];
#pragma unroll
    for (int k = 0; k < N_STATES; ++k) {
        FpR[k] = 0.f; FpI[k] = 0.f; FmR[k] = 0.f; FmI[k] = 0.f; Zs[k] = 0.f;
    }
    Zs[0] = 1.0f;

    const size_t pulseStride = (size_t)BATCH * EPT;             // floats per pulse
    const size_t blockOfs    = (size_t)blockIdx.x * CHUNK;      // block chunk within a pulse

#pragma unroll 1
    for (int p = 0; p < N_PULSES; ++p) {
        const float fa = flip[p];        // uniform -> scalar load
        const float ph = phases[p];      // uniform -> scalar load

        const float half = 0.5f * fa * B1;
        const float ca = __cosf(half), sa = __sinf(half);
        const float ca2 = ca * ca, sa2 = sa * sa, casa = ca * sa;
        const float czd = ca2 - sa2;
        const float ebr = __cosf(ph), ebi = __sinf(ph);          // exp(i*ph)
        const float e2r = ebr * ebr - ebi * ebi;                 // exp(2i*ph)
        const float e2i = 2.0f * ebr * ebi;

        // fused RF + relax + B0 + shift (Fp,Z roll +1 via carries; Fm roll -1 in place)
        float fpCr = 0.f, fpCi = 0.f, zC = 0.f;
#pragma unroll
        for (int k = 0; k < N_STATES; ++k) {
            const float fpr = FpR[k], fpi = FpI[k];
            const float fmr = FmR[k], fmi = FmI[k];
            const float z   = Zs[k];

            // RF pulse
            float nFpR = ca2 * fpr + sa2 * (fmr * e2r + fmi * e2i) - casa * z * ebi;
            float nFpI = ca2 * fpi + sa2 * (fmr * e2i - fmi * e2r) + casa * z * ebr;
            float nFmR = sa2 * (fpr * e2r - fpi * e2i) + ca2 * fmr - casa * z * ebi;
            float nFmI = -sa2 * (fpr * e2i + fpi * e2r) + ca2 * fmi - casa * z * ebr;
            float nZ   = casa * (fpi * ebr - fpr * ebi - fmr * ebi - fmi * ebr) + czd * z;

            // relaxation (+ regrowth on state 0, before shift)
            nFpR *= E2; nFpI *= E2; nFmR *= E2; nFmI *= E2;
            nZ = (k == 0) ? (nZ * E1 + regrow) : (nZ * E1);

            // B0 precession: Fp *= exp(i*phi), Fm *= exp(-i*phi)
            const float pr = nFpR * b0r - nFpI * b0i;
            const float pi_ = nFpR * b0i + nFpI * b0r;
            const float mr = nFmR * b0r + nFmI * b0i;
            const float mi = nFmI * b0r - nFmR * b0i;

            // EPG shift
            FpR[k] = fpCr; FpI[k] = fpCi; Zs[k] = zC;   // roll +1 (index 0 <- 0)
            fpCr = pr; fpCi = pi_; zC = nZ;
            if (k >= 1) { FmR[k - 1] = mr; FmI[k - 1] = mi; }   // roll -1
        }
        FmR[N_STATES - 1] = 0.f;
        FmI[N_STATES - 1] = 0.f;

        // ---- stage post-shift state into LDS, drain via async DMA ----
        const int buf = p & 1;
#if defined(EPG_ASYNC)
        epg_wait_async_le27();            // buffer 'buf' free (stores complete in order)
#endif
        {
            float* dst = &lds[buf][t * EPT];   // stride 105 dwords -> conflict-free banks
#pragma unroll
            for (int k = 0; k < N_STATES; ++k) {
                dst[k * 5 + 0] = FpR[k];
                dst[k * 5 + 1] = FpI[k];
                dst[k * 5 + 2] = FmR[k];
                dst[k * 5 + 3] = FmI[k];
                dst[k * 5 + 4] = Zs[k];
            }
        }
        epg_wait_dscnt0();                // LDS writes committed before async engine reads

        float* gdst = out + (size_t)p * pulseStride + blockOfs;   // 16B-aligned contiguous chunk
#if defined(EPG_ASYNC)
#pragma unroll 1
        for (int i = t; i < CHUNK4; i += TPB) {
            __builtin_amdgcn_global_store_async_from_lds_b128(
                (epg_gptr)(void*)(gdst + (size_t)i * 4),
                (epg_lptr)(void*)(&lds[buf][i * 4]),
                0, 0);
        }
#else
#pragma unroll 1
        for (int i = t; i < CHUNK4; i += TPB) {
            const float4 v = *reinterpret_cast<const float4*>(&lds[buf][i * 4]);
            *reinterpret_cast<float4*>(gdst + (size_t)i * 4) = v;
        }
#endif
    }
    // S_ENDPGM performs an implicit wait-idle (drains ASYNCcnt)
}

extern "C" void kernel_launch(void* const* d_in, const int* in_sizes, int n_in,
                              void* d_out, int out_size, void* d_ws, size_t ws_size,
                              hipStream_t stream) {
    (void)in_sizes; (void)n_in; (void)out_size; (void)d_ws; (void)ws_size;
    const float* flip   = (const float*)d_in[0];
    const float* phases = (const float*)d_in[1];
    const float* T1     = (const float*)d_in[2];
    const float* T2     = (const float*)d_in[3];
    const float* B0     = (const float*)d_in[4];
    const float* B1     = (const float*)d_in[5];
    const int*   TR     = (const int*)d_in[6];
    // d_in[7] = TE, unused by the reference forward
    float* out = (float*)d_out;

    dim3 grid(BATCH / TPB);
    dim3 block(TPB);
    epg_sim_kernel<<<grid, block, 0, stream>>>(flip, phases, T1, T2, B0, B1, TR, out);
}